// PointNext_82403242541244
// MI455X (gfx1250) — compile-verified
//
#include <hip/hip_runtime.h>
#include <math.h>

// Problem constants (match reference)
#define NPTS 65536
#define CCH  64
#define KNB  16
#define DBLK 2
#define EPSV 1e-5f

typedef float v2f __attribute__((ext_vector_type(2)));
typedef float v8f __attribute__((ext_vector_type(8)));

// =====================================================================
// GEMM: out[n,c] = relu( s[c]*acc[n,c] + t[c] (+ ident[n,c]) )
//   acc = xin @ W^T  (W is [64,64] row-major, c_out x c_in)
//   optional input pre-transform: x <- relu(preS[c]*x + preT[c])
// 256 threads = 8 waves; each wave computes a 16x64 output tile.
// kk-outer / nt-inner ordering keeps 4 independent WMMA accumulation
// chains in flight to hide XDL latency.
// =====================================================================
__global__ __launch_bounds__(256) void pn_gemm_rows(
    const float* __restrict__ xin, const float* __restrict__ W,
    float* __restrict__ out, const float* __restrict__ ident,
    const float* __restrict__ preS, const float* __restrict__ preT,
    const float* __restrict__ s, const float* __restrict__ t)
{
  __shared__ float wlds[64 * 64];
  {
    const float4* Wv = (const float4*)W;
    float4* Lv = (float4*)wlds;
    int tid = threadIdx.x;
#pragma unroll
    for (int i = 0; i < 4; ++i) Lv[tid + i * 256] = Wv[tid + i * 256];
  }
  __syncthreads();

  const int lane = threadIdx.x & 31;
  const int wave = threadIdx.x >> 5;
  const int m = lane & 15;       // M (row) for A, N (col) for B
  const int h = lane >> 4;       // lane-half selects K pair
  const int rowBase = blockIdx.x * 128 + wave * 16;
  const int row = rowBase + m;

  // A fragments: a[kk] = X[row, 4kk+2h .. 4kk+2h+1]
  v2f a[16];
  const float* xr = xin + (size_t)row * 64;
  const bool pre = (preS != nullptr);
#pragma unroll
  for (int kk = 0; kk < 16; ++kk) {
    const int c0 = 4 * kk + 2 * h;
    float x0 = xr[c0], x1 = xr[c0 + 1];
    if (pre) {
      x0 = fmaxf(fmaf(preS[c0], x0, preT[c0]), 0.0f);
      x1 = fmaxf(fmaf(preS[c0 + 1], x1, preT[c0 + 1]), 0.0f);
    }
    a[kk].x = x0; a[kk].y = x1;
  }

  v8f acc[4];
#pragma unroll
  for (int nt = 0; nt < 4; ++nt) acc[nt] = (v8f){};

#pragma unroll
  for (int kk = 0; kk < 16; ++kk) {
    const int c0 = 4 * kk + 2 * h;
#pragma unroll
    for (int nt = 0; nt < 4; ++nt) {
      const int col = nt * 16 + m;
      v2f b;
      b.x = wlds[col * 64 + c0];
      b.y = wlds[col * 64 + c0 + 1];
      acc[nt] = __builtin_amdgcn_wmma_f32_16x16x4_f32(
          false, a[kk], false, b, (short)0, acc[nt], false, false);
    }
  }

#pragma unroll
  for (int nt = 0; nt < 4; ++nt) {
    const int col = nt * 16 + m;
    const float sc = s[col], tc = t[col];
#pragma unroll
    for (int r = 0; r < 8; ++r) {
      const int orow = rowBase + r + 8 * h;
      float v = fmaf(acc[nt][r], sc, tc);
      if (ident) v += ident[(size_t)orow * 64 + col];
      out[(size_t)orow * 64 + col] = fmaxf(v, 0.0f);
    }
  }
}

// =====================================================================
// (Weighted) moment matrices:  M1[c]    += sum_r w_r * f(x[r,c])
//                              M2[c1,c2]+= sum_r w_r * f(x[r,c1])*f(x[r,c2])
// f = optional relu-affine pre-transform. M2 done as WMMA SYRK:
// A[m][k] = w * x[r0+k, c1b+m],  B[k][n] = x[r0+k, c2b+n].
// 8 waves x 2 adjacent c2-tiles (shared A fragment); row loop unrolled
// by 8 into two accumulator sets -> 4 independent WMMA chains.
// =====================================================================
__device__ __forceinline__ void pn_mom_step(
    const float* __restrict__ xin, const float* __restrict__ wgt, bool pre,
    float aS, float aT, float bS0, float bT0, float bS1, float bT1,
    int ra, int c1m, int c20m, int c21m, v8f& acc0, v8f& acc1)
{
  const float* x0 = xin + (size_t)ra * 64;
  const float* x1 = xin + (size_t)(ra + 1) * 64;
  const float w0 = wgt ? wgt[ra] : 1.0f;
  const float w1 = wgt ? wgt[ra + 1] : 1.0f;

  float a0 = x0[c1m],  a1 = x1[c1m];
  float p0 = x0[c20m], p1 = x1[c20m];
  float q0 = x0[c21m], q1 = x1[c21m];
  if (pre) {
    a0 = fmaxf(fmaf(aS, a0, aT), 0.f);   a1 = fmaxf(fmaf(aS, a1, aT), 0.f);
    p0 = fmaxf(fmaf(bS0, p0, bT0), 0.f); p1 = fmaxf(fmaf(bS0, p1, bT0), 0.f);
    q0 = fmaxf(fmaf(bS1, q0, bT1), 0.f); q1 = fmaxf(fmaf(bS1, q1, bT1), 0.f);
  }
  v2f A;  A.x = w0 * a0; A.y = w1 * a1;
  v2f B0; B0.x = p0;     B0.y = p1;
  v2f B1; B1.x = q0;     B1.y = q1;
  acc0 = __builtin_amdgcn_wmma_f32_16x16x4_f32(false, A, false, B0,
                                               (short)0, acc0, false, false);
  acc1 = __builtin_amdgcn_wmma_f32_16x16x4_f32(false, A, false, B1,
                                               (short)0, acc1, false, false);
}

__global__ __launch_bounds__(256) void pn_moments(
    const float* __restrict__ xin, const float* __restrict__ wgt,
    const float* __restrict__ preS, const float* __restrict__ preT,
    float* __restrict__ M1, float* __restrict__ M2, int rowsPerWG)
{
  const int lane = threadIdx.x & 31;
  const int wave = threadIdx.x >> 5;
  const int m = lane & 15;
  const int h = lane >> 4;
  const int r0 = blockIdx.x * rowsPerWG;

  const int t0 = wave * 2;
  const int c1b = (t0 >> 2) * 16;
  const int c2b0 = (t0 & 3) * 16;
  const int c2b1 = c2b0 + 16;

  const bool pre = (preS != nullptr);
  float aS = 1.f, aT = 0.f, bS0 = 1.f, bT0 = 0.f, bS1 = 1.f, bT1 = 0.f;
  if (pre) {
    aS = preS[c1b + m];   aT = preT[c1b + m];
    bS0 = preS[c2b0 + m]; bT0 = preT[c2b0 + m];
    bS1 = preS[c2b1 + m]; bT1 = preT[c2b1 + m];
  }

  v8f acc0 = {}, acc1 = {}, acc2 = {}, acc3 = {};
  for (int r = r0; r < r0 + rowsPerWG; r += 8) {
    pn_mom_step(xin, wgt, pre, aS, aT, bS0, bT0, bS1, bT1,
                r + 2 * h,     c1b + m, c2b0 + m, c2b1 + m, acc0, acc1);
    pn_mom_step(xin, wgt, pre, aS, aT, bS0, bT0, bS1, bT1,
                r + 4 + 2 * h, c1b + m, c2b0 + m, c2b1 + m, acc2, acc3);
  }
#pragma unroll
  for (int r = 0; r < 8; ++r) {
    const int c1 = c1b + r + 8 * h;
    atomicAdd(&M2[c1 * 64 + c2b0 + m], acc0[r] + acc2[r]);
    atomicAdd(&M2[c1 * 64 + c2b1 + m], acc1[r] + acc3[r]);
  }

  // M1: per-channel weighted sums (strided over 4 row-groups)
  const int c = threadIdx.x & 63;
  const int g = threadIdx.x >> 6;
  float s1 = 0.0f;
  float cS = 1.f, cT = 0.f;
  if (pre) { cS = preS[c]; cT = preT[c]; }
  for (int r = r0 + g; r < r0 + rowsPerWG; r += 4) {
    float x = xin[(size_t)r * 64 + c];
    if (pre) x = fmaxf(fmaf(cS, x, cT), 0.f);
    s1 += (wgt ? wgt[r] : 1.0f) * x;
  }
  atomicAdd(&M1[c], s1);
}

// =====================================================================
// Derive per-channel BN affine (scale, shift) for y = x@W^T + bias,
// from input moments:  E[y_c]  = (w_c.M1)/R + b_c
//                      E[y_c^2]= w_c^T M2 w_c /R + 2 b_c (w_c.M1)/R + b_c^2
// Epilogue form: bn(y) = scale*acc + shift (bias folded into shift).
// =====================================================================
__global__ __launch_bounds__(64) void pn_stats_from_moments(
    const float* __restrict__ W, const float* __restrict__ bias,
    const float* __restrict__ g, const float* __restrict__ bbn,
    const float* __restrict__ M1, const float* __restrict__ M2,
    float R, float* __restrict__ scale, float* __restrict__ shift)
{
  __shared__ float m2s[4096];
  __shared__ float m1s[64];
  for (int i = threadIdx.x; i < 4096; i += 64) m2s[i] = M2[i];
  if (threadIdx.x < 64) m1s[threadIdx.x] = M1[threadIdx.x];
  __syncthreads();

  const int c = threadIdx.x;
  const float* wr = W + c * 64;
  float wreg[64];
  float dot1 = 0.0f;
#pragma unroll
  for (int j = 0; j < 64; ++j) { wreg[j] = wr[j]; dot1 += wreg[j] * m1s[j]; }
  float quad = 0.0f;
  for (int i = 0; i < 64; ++i) {
    float acc = 0.0f;
#pragma unroll
    for (int j = 0; j < 64; ++j) acc += m2s[i * 64 + j] * wreg[j];
    quad += wreg[i] * acc;
  }
  const float bc = bias ? bias[c] : 0.0f;
  const float mean = dot1 / R + bc;
  const float ex2 = quad / R + 2.0f * bc * (dot1 / R) + bc * bc;
  const float var = ex2 - mean * mean;
  const float S = g[c] * rsqrtf(var + EPSV);
  scale[c] = S;
  shift[c] = fmaf(-mean, S, bbn[c]) + S * bc;
}

// Direct column stats (for BN after the k-max, which is not row-wise)
__global__ __launch_bounds__(256) void pn_colstats(
    const float* __restrict__ x, float* __restrict__ sum,
    float* __restrict__ sumsq, int rowsPerWG)
{
  const int c = threadIdx.x & 63;
  const int g = threadIdx.x >> 6;
  const int r0 = blockIdx.x * rowsPerWG;
  float s = 0.0f, q = 0.0f;
  for (int r = r0 + g; r < r0 + rowsPerWG; r += 4) {
    const float v = x[(size_t)r * 64 + c];
    s += v; q = fmaf(v, v, q);
  }
  atomicAdd(&sum[c], s);
  atomicAdd(&sumsq[c], q);
}

__global__ __launch_bounds__(64) void pn_finalize_colstats(
    const float* __restrict__ sum, const float* __restrict__ sumsq,
    const float* __restrict__ g, const float* __restrict__ b,
    float R, float* __restrict__ scale, float* __restrict__ shift)
{
  const int c = threadIdx.x;
  const float mean = sum[c] / R;
  const float var = sumsq[c] / R - mean * mean;
  const float S = g[c] * rsqrtf(var + EPSV);
  scale[c] = S;
  shift[c] = fmaf(-mean, S, b[c]);
}

// Histogram of neighbor indices (multiplicity of each unique row)
__global__ __launch_bounds__(256) void pn_hist(
    const int* __restrict__ ref, float* __restrict__ cnt, int total)
{
  const int i = blockIdx.x * blockDim.x + threadIdx.x;
  if (i < total) atomicAdd(&cnt[ref[i]], 1.0f);
}

// Gather + max over K neighbors: out[n,c] = max_k Z2[ref[n,k], c]
__global__ __launch_bounds__(256) void pn_max_gather(
    const float* __restrict__ Z2, const int* __restrict__ ref,
    float* __restrict__ out)
{
  const int c = threadIdx.x & 63;
  const int n = blockIdx.x * 4 + (threadIdx.x >> 6);
  const int* rr = ref + (size_t)n * KNB;
  float v = -INFINITY;
#pragma unroll
  for (int k = 0; k < KNB; ++k) {
    const int idx = rr[k];
    v = fmaxf(v, Z2[(size_t)idx * 64 + c]);
  }
  out[(size_t)n * 64 + c] = v;
}

// =====================================================================
extern "C" void kernel_launch(void* const* d_in, const int* in_sizes, int n_in,
                              void* d_out, int out_size, void* d_ws, size_t ws_size,
                              hipStream_t stream) {
  (void)in_sizes; (void)n_in; (void)out_size; (void)ws_size;
  const float* feat_in = (const float*)d_in[1];
  const int*   ref     = (const int*)d_in[2];
  const float* fc1_w   = (const float*)d_in[3];
  const float* bn1_g   = (const float*)d_in[4];
  const float* bn1_b   = (const float*)d_in[5];
  const float* la_w1   = (const float*)d_in[6];
  const float* la_b1   = (const float*)d_in[7];
  const float* la_bn1g = (const float*)d_in[8];
  const float* la_bn1b = (const float*)d_in[9];
  const float* la_w2   = (const float*)d_in[10];
  const float* la_b2   = (const float*)d_in[11];
  const float* la_bn2g = (const float*)d_in[12];
  const float* la_bn2b = (const float*)d_in[13];
  const float* fc3_w   = (const float*)d_in[14];
  const float* bn2_g   = (const float*)d_in[15];
  const float* bn2_b   = (const float*)d_in[16];
  const float* bn3_g   = (const float*)d_in[17];
  const float* bn3_b   = (const float*)d_in[18];

  // Workspace layout (floats): 3 big N*64 buffers + cnt + stats
  float* buf0 = (float*)d_ws;
  float* buf1 = buf0 + (size_t)NPTS * 64;
  float* buf2 = buf1 + (size_t)NPTS * 64;
  float* cnt  = buf2 + (size_t)NPTS * 64;
  float* M1   = cnt + NPTS;       // 64
  float* M2   = M1 + 64;          // 4096 (contiguous with M1 for memset)
  float* csum = M2 + 4096;        // 64
  float* csq  = csum + 64;        // 64 (contiguous)
  float* sc1 = csq + 64;   float* sh1 = sc1 + 64;
  float* sc2 = sh1 + 64;   float* sh2 = sc2 + 64;
  float* sc3 = sh2 + 64;   float* sh3 = sc3 + 64;
  float* sc4 = sh3 + 64;   float* sh4 = sc4 + 64;
  float* sc5 = sh4 + 64;   float* sh5 = sc5 + 64;

  const dim3 gemmGrid(NPTS / 128), b256(256), b64(64);
  const float Rn = (float)NPTS, Rnk = (float)NPTS * (float)KNB;

  // Neighbor multiplicities (ref is constant across blocks)
  hipMemsetAsync(cnt, 0, NPTS * sizeof(float), stream);
  pn_hist<<<(NPTS * KNB + 255) / 256, b256, 0, stream>>>(ref, cnt, NPTS * KNB);

  for (int d = 0; d < DBLK; ++d) {
    const float* featIn  = (d == 0) ? feat_in : buf2;
    float*       featOut = (d == 0) ? buf2 : (float*)d_out;
    float* X  = buf0;   // relu(bn1(feat@W1^T))
    float* Z  = buf1;   // relu(bn(la1)) on unique rows
    float* Z2 = buf0;   // relu(bn(la2)) on unique rows (reuse)
    float* Mx = buf1;   // max over K (reuse)
    const float* W1 = fc1_w + d * 4096;
    const float* Wa = la_w1 + d * 4096;
    const float* Wb = la_w2 + d * 4096;
    const float* W3 = fc3_w + d * 4096;

    // ---- fc1 + bn1 + relu (bn1 stats via moments of featIn through W1)
    hipMemsetAsync(M1, 0, (64 + 4096) * sizeof(float), stream);
    pn_moments<<<64, b256, 0, stream>>>(featIn, nullptr, nullptr, nullptr,
                                        M1, M2, NPTS / 64);
    pn_stats_from_moments<<<1, b64, 0, stream>>>(W1, nullptr, bn1_g + d * 64,
        bn1_b + d * 64, M1, M2, Rn, sc1, sh1);
    pn_gemm_rows<<<gemmGrid, b256, 0, stream>>>(featIn, W1, X, nullptr,
        nullptr, nullptr, sc1, sh1);

    // ---- LA layer 1 on unique rows; BN stats are cnt-weighted (R = N*K)
    hipMemsetAsync(M1, 0, (64 + 4096) * sizeof(float), stream);
    pn_moments<<<64, b256, 0, stream>>>(X, cnt, nullptr, nullptr, M1, M2,
                                        NPTS / 64);
    pn_stats_from_moments<<<1, b64, 0, stream>>>(Wa, la_b1 + d * 64,
        la_bn1g + d * 64, la_bn1b + d * 64, M1, M2, Rnk, sc2, sh2);
    pn_gemm_rows<<<gemmGrid, b256, 0, stream>>>(X, Wa, Z, nullptr,
        nullptr, nullptr, sc2, sh2);

    // ---- LA layer 2 on unique rows
    hipMemsetAsync(M1, 0, (64 + 4096) * sizeof(float), stream);
    pn_moments<<<64, b256, 0, stream>>>(Z, cnt, nullptr, nullptr, M1, M2,
                                        NPTS / 64);
    pn_stats_from_moments<<<1, b64, 0, stream>>>(Wb, la_b2 + d * 64,
        la_bn2g + d * 64, la_bn2b + d * 64, M1, M2, Rnk, sc3, sh3);
    pn_gemm_rows<<<gemmGrid, b256, 0, stream>>>(Z, Wb, Z2, nullptr,
        nullptr, nullptr, sc3, sh3);

    // ---- gather + max over K
    pn_max_gather<<<NPTS / 4, b256, 0, stream>>>(Z2, ref, Mx);

    // ---- bn2 (direct column stats over the maxed rows)
    hipMemsetAsync(csum, 0, 128 * sizeof(float), stream);
    pn_colstats<<<64, b256, 0, stream>>>(Mx, csum, csq, NPTS / 64);
    pn_finalize_colstats<<<1, b64, 0, stream>>>(csum, csq, bn2_g + d * 64,
        bn2_b + d * 64, Rn, sc4, sh4);

    // ---- bn3 stats via moments of u = relu(sc4*Mx + sh4) through W3
    hipMemsetAsync(M1, 0, (64 + 4096) * sizeof(float), stream);
    pn_moments<<<64, b256, 0, stream>>>(Mx, nullptr, sc4, sh4, M1, M2,
                                        NPTS / 64);
    pn_stats_from_moments<<<1, b64, 0, stream>>>(W3, nullptr, bn3_g + d * 64,
        bn3_b + d * 64, M1, M2, Rn, sc5, sh5);

    // ---- fc3 (+bn3) with fused input transform and residual + relu
    pn_gemm_rows<<<gemmGrid, b256, 0, stream>>>(Mx, W3, featOut, featIn,
        sc4, sh4, sc5, sh5);
  }
}